// RNN_35244501631277
// MI455X (gfx1250) — compile-verified
//
#include <hip/hip_runtime.h>
#include <math.h>

#define H        2048
#define NSTEPS   119
#define BATCH    16384
#define NDEPTH   2
#define CHAIN_NWG     16      // workgroups per chain (2048 cols / 128 cols-per-WG)
#define CHAIN_THREADS 256     // 8 waves (wave32)
#define COLS_PER_WAVE 16
#define COLS_PER_WG   128

typedef __attribute__((ext_vector_type(2))) float v2f;
typedef __attribute__((ext_vector_type(8))) float v8f;
typedef __attribute__((__vector_size__(4 * sizeof(int)))) int v4i;

#if defined(__has_builtin)
#if __has_builtin(__builtin_amdgcn_global_load_async_to_lds_b128)
#define USE_ASYNC_LDS 1
#endif
#if __has_builtin(__builtin_amdgcn_s_wait_asynccnt)
#define HAVE_WAIT_ASYNCCNT_BUILTIN 1
#endif
#endif

__device__ __forceinline__ void wait_asynccnt0() {
#ifdef HAVE_WAIT_ASYNCCNT_BUILTIN
  __builtin_amdgcn_s_wait_asynccnt(0);
#else
  asm volatile("s_wait_asynccnt 0" ::: "memory");
#endif
}

// ---------------------------------------------------------------------------
// Stage the H-float input vector (8 KB) from global into LDS.
// CDNA5 path: async global->LDS copy tracked by ASYNCcnt.
// ---------------------------------------------------------------------------
__device__ __forceinline__ void stage_vec_to_lds(const float* __restrict__ g,
                                                 float* lds) {
#ifdef USE_ASYNC_LDS
  typedef __attribute__((address_space(1))) v4i* gas_v4i;
  typedef __attribute__((address_space(3))) v4i* las_v4i;
  for (int i = threadIdx.x; i < H / 4; i += CHAIN_THREADS) {
    __builtin_amdgcn_global_load_async_to_lds_b128(
        (gas_v4i)(g + 4 * i), (las_v4i)(lds + 4 * i), 0, 0);
  }
  wait_asynccnt0();
#else
  for (int i = threadIdx.x; i < H / 4; i += CHAIN_THREADS) {
    ((float4*)lds)[i] = ((const float4*)g)[i];
  }
#endif
  __syncthreads();
}

// ---------------------------------------------------------------------------
// Grid-wide barrier over the CHAIN_NWG workgroups of one chain
// (atomic sense-reversal; all 32 WGs are co-resident on MI455X).
// ---------------------------------------------------------------------------
__device__ __forceinline__ void grid_barrier(int* counter, int* gen) {
  __threadfence();          // release this WG's stores device-wide
  __syncthreads();
  if (threadIdx.x == 0) {
    int g = __hip_atomic_load(gen, __ATOMIC_RELAXED, __HIP_MEMORY_SCOPE_AGENT);
    int prev = __hip_atomic_fetch_add(counter, 1, __ATOMIC_ACQ_REL,
                                      __HIP_MEMORY_SCOPE_AGENT);
    if (prev == CHAIN_NWG - 1) {
      __hip_atomic_store(counter, 0, __ATOMIC_RELAXED, __HIP_MEMORY_SCOPE_AGENT);
      __hip_atomic_fetch_add(gen, 1, __ATOMIC_RELEASE, __HIP_MEMORY_SCOPE_AGENT);
    } else {
      while (__hip_atomic_load(gen, __ATOMIC_ACQUIRE,
                               __HIP_MEMORY_SCOPE_AGENT) == g) {
        __builtin_amdgcn_s_sleep(2);
      }
    }
  }
  __syncthreads();
  __threadfence();          // acquire: invalidate stale WGP cache lines
}

// ---------------------------------------------------------------------------
// One matvec: out[j] = silu(in . W[:,j] + bias[j]) for this wave's 16 columns,
// via V_WMMA_F32_16X16X4_F32 with the input vector broadcast across all 16
// A-rows (every D row then equals the matvec result).
//
// Fragment layout assumptions (ISA 7.12.2, 32-bit operands):
//   A 16x4 : VGPR0 = K{0,2} (lanes 0-15 / 16-31), VGPR1 = K{1,3}
//   B 4x16 : VGPR0 = rows K{0,2} (N=lane&15),     VGPR1 = rows K{1,3}
//   D      : VGPR0, lanes 0-15 = row 0, N = lane
// ---------------------------------------------------------------------------
__device__ __forceinline__ void matvec_silu(
    const float* __restrict__ in_vec,   // H floats (global)
    const float* __restrict__ Wd,       // H x H row-major
    const float* __restrict__ bias,     // H floats
    float* __restrict__ out_vec,        // H floats
    float* __restrict__ tab_row,        // H floats or nullptr
    bool accumulate, float* lds, int colbase)
{
  stage_vec_to_lds(in_vec, lds);

  const int lane = threadIdx.x & 31;
  const int kofs = (lane >> 4) << 1;    // 0 for lanes 0-15, 2 for lanes 16-31
  const int col  = lane & 15;
  const int j    = colbase + col;

  v8f acc = {0.f, 0.f, 0.f, 0.f, 0.f, 0.f, 0.f, 0.f};
  const float* __restrict__ wp0 = Wd + j;

#pragma unroll 8
  for (int k = 0; k < H; k += 4) {
    float2 pv = *(const float2*)(lds + k + kofs);       // ds_load_b64 broadcast
    v2f a; a.x = pv.x; a.y = pv.y;
    const float* wp = wp0 + (size_t)(k + kofs) * H;
    v2f b; b.x = wp[0]; b.y = wp[H];                    // coalesced 64B rows
    acc = __builtin_amdgcn_wmma_f32_16x16x4_f32(
        false, a, false, b, (short)0, acc, false, false);
  }

  if (lane < 16) {
    float v = acc[0] + bias[j];
    float s = v / (1.0f + __expf(-v));                  // silu
    if (accumulate) {
      float nv = out_vec[j] + s;
      out_vec[j] = nv;
      if (tab_row) tab_row[j] = nv;
    } else {
      out_vec[j] = s;
    }
  }
}

// ---------------------------------------------------------------------------
// Persistent kernel: runs both 119-step recurrent chains.
// blockIdx [0,16) -> proton chain, [16,32) -> neutron chain.
// ---------------------------------------------------------------------------
__global__ void __launch_bounds__(CHAIN_THREADS)
chain_kernel(const float* __restrict__ pemb, const float* __restrict__ nemb,
             const float* __restrict__ pW1, const float* __restrict__ pb1,
             const float* __restrict__ pW2, const float* __restrict__ pb2,
             const float* __restrict__ nW1, const float* __restrict__ nb1,
             const float* __restrict__ nW2, const float* __restrict__ nb2,
             float* __restrict__ ptab, float* __restrict__ ntab,
             float* __restrict__ pvec, float* __restrict__ phv,
             float* __restrict__ nvec, float* __restrict__ nhv,
             int* __restrict__ bar)
{
  __shared__ float s_in[H];

  const int chain = blockIdx.x / CHAIN_NWG;
  const int wg    = blockIdx.x % CHAIN_NWG;
  const float* emb = chain ? nemb : pemb;
  const float* W1  = chain ? nW1 : pW1;
  const float* B1  = chain ? nb1 : pb1;
  const float* W2  = chain ? nW2 : pW2;
  const float* B2  = chain ? nb2 : pb2;
  float* tab = chain ? ntab : ptab;
  float* vec = chain ? nvec : pvec;
  float* hv  = chain ? nhv  : phv;
  int* counter = bar + 2 * chain;
  int* gen     = counter + 1;

  const int wave    = threadIdx.x >> 5;
  const int colbase = wg * COLS_PER_WG + wave * COLS_PER_WAVE;

  // p0 = emb
  if (threadIdx.x < COLS_PER_WG)
    vec[wg * COLS_PER_WG + threadIdx.x] = emb[wg * COLS_PER_WG + threadIdx.x];
  grid_barrier(counter, gen);

  for (int t = 0; t < NSTEPS; ++t) {
    for (int d = 0; d < NDEPTH; ++d) {
      // h = silu(p @ W1[d] + b1[d])
      matvec_silu(vec, W1 + (size_t)d * H * H, B1 + (size_t)d * H,
                  hv, nullptr, false, s_in, colbase);
      grid_barrier(counter, gen);
      // p += silu(h @ W2[d] + b2[d]);  record p into table on last depth
      matvec_silu(hv, W2 + (size_t)d * H * H, B2 + (size_t)d * H,
                  vec, (d == NDEPTH - 1) ? (tab + (size_t)t * H) : nullptr,
                  true, s_in, colbase);
      grid_barrier(counter, gen);
    }
  }
}

// ---------------------------------------------------------------------------
// pdot[t] = dot(ptab[t], rW[0:H]);  ndot[t] = dot(ntab[t], rW[H:2H])
// ---------------------------------------------------------------------------
__global__ void __launch_bounds__(256)
dot_kernel(const float* __restrict__ ptab, const float* __restrict__ ntab,
           const float* __restrict__ rW,
           float* __restrict__ pdot, float* __restrict__ ndot)
{
  const int which = blockIdx.x / NSTEPS;
  const int t     = blockIdx.x % NSTEPS;
  const float* __restrict__ row = (which ? ntab : ptab) + (size_t)t * H;
  const float* __restrict__ w   = rW + which * H;   // rW is (2H, 1)

  float s = 0.f;
  for (int i = threadIdx.x; i < H; i += 256) s = fmaf(row[i], w[i], s);
  for (int off = 16; off > 0; off >>= 1) s += __shfl_down(s, off, 32);

  __shared__ float red[8];
  const int wv = threadIdx.x >> 5, ln = threadIdx.x & 31;
  if (ln == 0) red[wv] = s;
  __syncthreads();
  if (threadIdx.x == 0) {
    float tot = 0.f;
    for (int i = 0; i < 8; ++i) tot += red[i];
    (which ? ndot : pdot)[t] = tot;
  }
}

// ---------------------------------------------------------------------------
// out[b] = sigmoid(pdot[x[b,0]] + ndot[x[b,1]] + rb)
// ---------------------------------------------------------------------------
__global__ void __launch_bounds__(256)
out_kernel(const int* __restrict__ x,
           const float* __restrict__ pdot, const float* __restrict__ ndot,
           const float* __restrict__ rb, float* __restrict__ out, int n)
{
  const int b = blockIdx.x * blockDim.x + threadIdx.x;
  if (b < n) {
    float v = pdot[x[2 * b]] + ndot[x[2 * b + 1]] + rb[0];
    out[b] = 1.0f / (1.0f + __expf(-v));
  }
}

__global__ void init_kernel(int* __restrict__ bar) {
  if (threadIdx.x < 4) bar[threadIdx.x] = 0;
}

// ---------------------------------------------------------------------------
// Workspace layout (floats): ptab | ntab | pvec | phv | nvec | nhv | pdot | ndot | bar
// ---------------------------------------------------------------------------
extern "C" void kernel_launch(void* const* d_in, const int* in_sizes, int n_in,
                              void* d_out, int out_size, void* d_ws, size_t ws_size,
                              hipStream_t stream) {
  const int*   x    = (const int*)d_in[0];
  const float* pemb = (const float*)d_in[1];
  const float* nemb = (const float*)d_in[2];
  const float* pW1  = (const float*)d_in[3];
  const float* pb1  = (const float*)d_in[4];
  const float* pW2  = (const float*)d_in[5];
  const float* pb2  = (const float*)d_in[6];
  const float* nW1  = (const float*)d_in[7];
  const float* nb1  = (const float*)d_in[8];
  const float* nW2  = (const float*)d_in[9];
  const float* nb2  = (const float*)d_in[10];
  const float* rW   = (const float*)d_in[11];
  const float* rb   = (const float*)d_in[12];
  float* out = (float*)d_out;

  float* ws   = (float*)d_ws;
  float* ptab = ws;
  float* ntab = ptab + (size_t)NSTEPS * H;
  float* pvec = ntab + (size_t)NSTEPS * H;
  float* phv  = pvec + H;
  float* nvec = phv + H;
  float* nhv  = nvec + H;
  float* pdot = nhv + H;
  float* ndot = pdot + 128;
  int*   bar  = (int*)(ndot + 128);

  init_kernel<<<1, 64, 0, stream>>>(bar);
  chain_kernel<<<2 * CHAIN_NWG, CHAIN_THREADS, 0, stream>>>(
      pemb, nemb, pW1, pb1, pW2, pb2, nW1, nb1, nW2, nb2,
      ptab, ntab, pvec, phv, nvec, nhv, bar);
  dot_kernel<<<2 * NSTEPS, 256, 0, stream>>>(ptab, ntab, rW, pdot, ndot);
  out_kernel<<<(BATCH + 255) / 256, 256, 0, stream>>>(x, pdot, ndot, rb, out, BATCH);
}